// RnCLoss_trans_mug_mix_79147657331292
// MI455X (gfx1250) — compile-verified
//
#include <hip/hip_runtime.h>
#include <math.h>

#define N     384
#define DF    128
#define NM1   383   // N-1

typedef __attribute__((ext_vector_type(2))) float v2f;
typedef __attribute__((ext_vector_type(8))) float v8f;

// ---------------------------------------------------------------------------
// Kernel 1: G = F * F^T via V_WMMA_F32_16X16X4_F32 (exact fp32 matrix path).
// 24x24 tiles of 16x16; one tile per wave; 8 waves per 256-thread block.
// ---------------------------------------------------------------------------
__global__ __launch_bounds__(256) void gram_wmma_kernel(
    const float* __restrict__ F, float* __restrict__ G) {
  const int lane = threadIdx.x & 31;
  const int wave = threadIdx.x >> 5;
  const int tile = blockIdx.x * 8 + wave;   // 0..575
  const int tm = tile / 24;
  const int tn = tile % 24;
  const int m0 = tm * 16;
  const int n0 = tn * 16;
  const int half = lane >> 4;               // 0: lanes 0-15, 1: lanes 16-31
  const int lm   = lane & 15;

  // A-tile row for this lane (16x4 fp32 A layout: VGPR0=K0/K2, VGPR1=K1/K3)
  const float* arow = F + (m0 + lm) * DF + 2 * half;
  // B-tile (B = F^T): b[k][n] = F[n][k]  -> same addressing form with n0
  const float* brow = F + (n0 + lm) * DF + 2 * half;

  v8f acc = {};
#pragma unroll 4
  for (int k = 0; k < DF; k += 4) {
    v2f a = *(const v2f*)(arow + k);
    v2f b = *(const v2f*)(brow + k);
    acc = __builtin_amdgcn_wmma_f32_16x16x4_f32(
        /*neg_a=*/false, a, /*neg_b=*/false, b,
        /*c_mod=*/(short)0, acc, /*reuse_a=*/false, /*reuse_b=*/false);
  }

  // D layout: VGPR v -> M = m0 + v + 8*half, N = n0 + lm
#pragma unroll
  for (int v = 0; v < 8; ++v) {
    G[(m0 + v + 8 * half) * N + (n0 + lm)] = acc[v];
  }
}

// ---------------------------------------------------------------------------
// Kernel 2: one block per row i. Builds off-diagonal lo/eo/td/rd in LDS,
// then the O(n^2)-per-row dominance sums, block-reduces partial losses.
// ---------------------------------------------------------------------------
__device__ __forceinline__ float smooth_l1(float d) {
  float ad = fabsf(d);
  return (ad < 1.0f) ? (0.5f * d * d) : (ad - 0.5f);
}

__global__ __launch_bounds__(256) void row_kernel(
    const float* __restrict__ G,
    const float* __restrict__ rot_green,
    const float* __restrict__ rot_red,
    const float* __restrict__ trans,
    const int*   __restrict__ sym,
    float* __restrict__ partH, float* __restrict__ partT) {
  __shared__ float s_lo[NM1];
  __shared__ float s_eo[NM1];
  __shared__ float s_td[NM1];
  __shared__ float s_rd[NM1];
  __shared__ float s_rh[256];
  __shared__ float s_rt[256];

  const int i   = blockIdx.x;
  const int tid = threadIdx.x;

  const float ti0 = trans[i * 3 + 0];
  const float ti1 = trans[i * 3 + 1];
  const float ti2 = trans[i * 3 + 2];
  const float gi0 = rot_green[i * 3 + 0];
  const float gi1 = rot_green[i * 3 + 1];
  const float gi2 = rot_green[i * 3 + 2];
  const float ri0 = rot_red[i * 3 + 0];
  const float ri1 = rot_red[i * 3 + 1];
  const float ri2 = rot_red[i * 3 + 2];
  const float gin = fmaxf(sqrtf(gi0 * gi0 + gi1 * gi1 + gi2 * gi2), 1e-8f);
  const float rin = fmaxf(sqrtf(ri0 * ri0 + ri1 * ri1 + ri2 * ri2), 1e-8f);
  const bool  symi = (sym[i] == 1);
  const float Gii  = G[i * N + i];

  for (int c = tid; c < NM1; c += 256) {
    const int j = c + (c >= i);   // off-diagonal column mapping

    // trans smooth-L1, mean over 3 coords
    const float d0 = ti0 - trans[j * 3 + 0];
    const float d1 = ti1 - trans[j * 3 + 1];
    const float d2 = ti2 - trans[j * 3 + 2];
    const float td = (smooth_l1(d0) + smooth_l1(d1) + smooth_l1(d2)) * (1.0f / 3.0f);

    // rotation cosine diffs
    const float gj0 = rot_green[j * 3 + 0];
    const float gj1 = rot_green[j * 3 + 1];
    const float gj2 = rot_green[j * 3 + 2];
    const float gjn = fmaxf(sqrtf(gj0 * gj0 + gj1 * gj1 + gj2 * gj2), 1e-8f);
    const float g   = 1.0f - (gi0 * gj0 + gi1 * gj1 + gi2 * gj2) / (gin * gjn);

    const float rj0 = rot_red[j * 3 + 0];
    const float rj1 = rot_red[j * 3 + 1];
    const float rj2 = rot_red[j * 3 + 2];
    const float rjn = fmaxf(sqrtf(rj0 * rj0 + rj1 * rj1 + rj2 * rj2), 1e-8f);
    const float r   = 1.0f - (ri0 * rj0 + ri1 * rj1 + ri2 * rj2) / (rin * rjn);

    const bool  ps = symi || (sym[j] == 1);
    const float rd = ps ? g : (g + r);

    // feature logits from Gram matrix; row-max is exactly 0 (diagonal)
    const float sq  = Gii + G[j * N + j] - 2.0f * G[i * N + j];
    const float nrm = (sq > 0.0f) ? sqrtf(sq) : 0.0f;
    const float lo  = -nrm * 0.5f;   // /TEMPERATURE (=2)

    s_lo[c] = lo;
    s_eo[c] = expf(lo);
    s_td[c] = td;
    s_rd[c] = rd;
  }
  __syncthreads();

  float sH = 0.0f, sT = 0.0f;
  for (int c = tid; c < NM1; c += 256) {
    const float tdj = s_td[c];
    const float rdj = s_rd[c];
    float dT = 0.0f, dH = 0.0f;
    for (int k = 0; k < NM1; ++k) {           // LDS broadcast reads
      const float e  = s_eo[k];
      const bool  mt = (s_td[k] >= tdj);      // trans_mask: td[j] <= td[k]
      dT += mt ? e : 0.0f;
      dH += (mt && (s_rd[k] >= rdj)) ? e : 0.0f;
    }
    sT += s_lo[c] - logf(dT + 1e-7f);
    sH += s_lo[c] - logf(dH + 1e-7f);
  }

  s_rh[tid] = sH;
  s_rt[tid] = sT;
  __syncthreads();
  for (int s = 128; s > 0; s >>= 1) {
    if (tid < s) {
      s_rh[tid] += s_rh[tid + s];
      s_rt[tid] += s_rt[tid + s];
    }
    __syncthreads();
  }
  if (tid == 0) {
    partH[i] = s_rh[0];
    partT[i] = s_rt[0];
  }
}

// ---------------------------------------------------------------------------
// Kernel 3: final deterministic reduction -> scalar loss.
// ---------------------------------------------------------------------------
__global__ __launch_bounds__(256) void final_kernel(
    const float* __restrict__ partH, const float* __restrict__ partT,
    float* __restrict__ out) {
  __shared__ float s_rh[256];
  __shared__ float s_rt[256];
  const int tid = threadIdx.x;
  float sH = 0.0f, sT = 0.0f;
  for (int t = tid; t < N; t += 256) {
    sH += partH[t];
    sT += partT[t];
  }
  s_rh[tid] = sH;
  s_rt[tid] = sT;
  __syncthreads();
  for (int s = 128; s > 0; s >>= 1) {
    if (tid < s) {
      s_rh[tid] += s_rh[tid + s];
      s_rt[tid] += s_rt[tid + s];
    }
    __syncthreads();
  }
  if (tid == 0) {
    const float denom = (float)N * (float)(N - 1);
    // loss_hard + 0.5*loss_trans = (-sumH - 0.5*sumT)/denom
    out[0] = (-s_rh[0] - 0.5f * s_rt[0]) / denom;
  }
}

// ---------------------------------------------------------------------------
extern "C" void kernel_launch(void* const* d_in, const int* in_sizes, int n_in,
                              void* d_out, int out_size, void* d_ws, size_t ws_size,
                              hipStream_t stream) {
  (void)in_sizes; (void)n_in; (void)out_size; (void)ws_size;
  const float* F   = (const float*)d_in[0];  // trans_feat [384,128]
  const float* rg  = (const float*)d_in[1];  // rot_green  [384,3]
  const float* rr  = (const float*)d_in[2];  // rot_red    [384,3]
  const float* tr  = (const float*)d_in[3];  // trans      [384,3]
  const int*   sym = (const int*)  d_in[4];  // sym        [384,1]

  float* G     = (float*)d_ws;     // 384*384 floats
  float* partH = G + N * N;        // 384 floats
  float* partT = partH + N;        // 384 floats

  gram_wmma_kernel<<<72, 256, 0, stream>>>(F, G);                 // 576 tiles / 8 waves
  row_kernel<<<N, 256, 0, stream>>>(G, rg, rr, tr, sym, partH, partT);
  final_kernel<<<1, 256, 0, stream>>>(partH, partT, (float*)d_out);
}